// Backbone_47914655154300
// MI455X (gfx1250) — compile-verified
//
#include <hip/hip_runtime.h>
#include <cstdint>
#include <cmath>

// ---------------------------------------------------------------------------
// Types / CDNA5 primitives
// ---------------------------------------------------------------------------
typedef float v2f __attribute__((ext_vector_type(2)));
typedef float v8f __attribute__((ext_vector_type(8)));
typedef unsigned int u32x4 __attribute__((ext_vector_type(4)));
typedef int i32x4 __attribute__((ext_vector_type(4)));
typedef int i32x8 __attribute__((ext_vector_type(8)));

__device__ __forceinline__ v8f wmma_f32_16x16x4(v2f a, v2f b, v8f c) {
  // D = A(16x4 f32) * B(4x16 f32) + C(16x16 f32), wave32
  return __builtin_amdgcn_wmma_f32_16x16x4_f32(false, a, false, b, (short)0, c,
                                               false, false);
}

__device__ __forceinline__ float gelu_erf(float x) {
  return 0.5f * x * (1.0f + erff(x * 0.7071067811865475f));
}
__device__ __forceinline__ float sigmoid_f(float x) {
  return 1.0f / (1.0f + expf(-x));
}
__device__ __forceinline__ float act_apply(float v, int act) {
  if (act == 1) return gelu_erf(v);
  if (act == 2) return sigmoid_f(v);
  return v;
}

// Tensor Data Mover: async 2D tile (tile_w x tile_h f32 elems, row stride in
// elems) from global -> LDS.  D# built per CDNA5 ISA.
__device__ __forceinline__ void tdm_load_2d(unsigned lds_off, const float* gsrc,
                                            unsigned tile_w, unsigned tile_h,
                                            unsigned long long row_stride) {
  unsigned long long ga = (unsigned long long)gsrc;
  u32x4 g0;
  g0.x = 1u;                                   // count=1, user descriptor
  g0.y = lds_off;                              // lds_addr
  g0.z = (unsigned)ga;                         // global_addr[31:0]
  g0.w = (unsigned)((ga >> 32) & 0x01FFFFFFull) | (2u << 30);  // addr hi + type=2
  const unsigned td0 = 0x7FFFFFFFu, td1 = 0x7FFFFFFFu;  // huge dims: no OOB clip
  i32x8 g1;
  g1[0] = (int)(2u << 16);                                    // data_size=4B
  g1[1] = (int)((td0 & 0xFFFFu) << 16);                       // tensor_dim0 lo
  g1[2] = (int)(((td0 >> 16) & 0xFFFFu) | ((td1 & 0xFFFFu) << 16));
  g1[3] = (int)(((td1 >> 16) & 0xFFFFu) | ((tile_w & 0xFFFFu) << 16)); // tile_dim0
  g1[4] = (int)(tile_h & 0xFFFFu);                            // tile_dim1, dim2=0
  g1[5] = (int)(unsigned)(row_stride & 0xFFFFFFFFull);        // dim0_stride lo
  g1[6] = (int)(unsigned)((row_stride >> 32) & 0xFFFFull);    // dim0_stride hi
  g1[7] = 0;
  i32x4 z4 = {0, 0, 0, 0};
#if __clang_major__ >= 23
  i32x8 z8 = {0, 0, 0, 0, 0, 0, 0, 0};
  __builtin_amdgcn_tensor_load_to_lds(g0, g1, z4, z4, z8, 0);
#else
  __builtin_amdgcn_tensor_load_to_lds(g0, g1, z4, z4, 0);
#endif
}

// ---------------------------------------------------------------------------
// 1x1-conv implicit GEMM:  Y[b][Cout][HW] = W[Cout][Cin] * X[b][Cin][HW] + bias
// grid = (HW/128, ceil(Cout/16), B), block = 256 (8 waves).  Each wave: one
// 16(M) x 16(N) fp32 WMMA tile.  Activations staged to LDS by the TDM in
// double-buffered K-chunks of 32 channels; weights from global (L2-resident),
// loaded branchlessly (clamped row + 0/1 mask) so the K loop has no exec-mask
// traffic, with all 8 A-fragments of a chunk in flight before the WMMA chain.
// act: 0=none 1=gelu 2=sigmoid.  res: optional residual (same layout as Y).
// Requires Cin % 32 == 0, HW % 128 == 0 (holds for every call in this net).
// ---------------------------------------------------------------------------
__global__ __launch_bounds__(256) void gemm1x1_kernel(
    const float* __restrict__ X, const float* __restrict__ Wt,
    const float* __restrict__ bias, const float* __restrict__ res,
    float* __restrict__ Y, int Cin, int Cout, int HW, int act) {
  __shared__ float xs[2][32 * 128];
  const int b = blockIdx.z;
  const int m0 = blockIdx.y * 16;
  const int n0 = blockIdx.x * 128;
  const int lane = threadIdx.x & 31;
  const int wave = threadIdx.x >> 5;
  const int half = lane >> 4;   // K-split half per fp32 WMMA layout
  const int l16 = lane & 15;
  const float* Xb = X + (size_t)b * Cin * HW + n0;
  const unsigned ldsbase = (unsigned)(uintptr_t)(&xs[0][0]);
  const int nchunks = Cin >> 5;

  // Branchless A-row: clamp to a valid address, mask the value (only the
  // seg conv has Cout<16).
  const int row = m0 + l16;
  const int rowc = (row < Cout) ? row : (Cout - 1);
  const float wmask = (row < Cout) ? 1.f : 0.f;
  const float* wrow = Wt + (size_t)rowc * Cin + half * 2;

  if (wave == 0) tdm_load_2d(ldsbase, Xb, 128, 32, (unsigned long long)HW);

  v8f acc = {0.f, 0.f, 0.f, 0.f, 0.f, 0.f, 0.f, 0.f};
  for (int ch = 0; ch < nchunks; ++ch) {
    if (wave == 0) {
      if (ch + 1 < nchunks) {
        tdm_load_2d(ldsbase + (unsigned)(((ch + 1) & 1) * 32 * 128 * 4),
                    Xb + (size_t)(ch + 1) * 32 * HW, 128, 32,
                    (unsigned long long)HW);
        __builtin_amdgcn_s_wait_tensorcnt(1);   // chunk `ch` landed
      } else {
        __builtin_amdgcn_s_wait_tensorcnt(0);
      }
    }
    // Preload this chunk's 8 A-fragments (aligned float2 loads, all in
    // flight together); prefetch next chunk's weight row.
    v2f afrag[8];
#pragma unroll
    for (int k4 = 0; k4 < 8; ++k4) {
      const v2f w2 = *(const v2f*)(wrow + ch * 32 + k4 * 4);
      afrag[k4].x = w2.x * wmask;
      afrag[k4].y = w2.y * wmask;
    }
    if (ch + 1 < nchunks)
      __builtin_prefetch(wrow + (ch + 1) * 32, 0, 1);
    __syncthreads();
    const float* xls = &xs[ch & 1][0];
#pragma unroll
    for (int k4 = 0; k4 < 8; ++k4) {
      const int kl = k4 * 4 + half * 2;
      v2f bf;
      bf.x = xls[kl * 128 + wave * 16 + l16];
      bf.y = xls[(kl + 1) * 128 + wave * 16 + l16];
      acc = wmma_f32_16x16x4(afrag[k4], bf, acc);
    }
    __syncthreads();
  }

  const int col = n0 + wave * 16 + l16;
#pragma unroll
  for (int r = 0; r < 8; ++r) {
    const int orow = m0 + r + half * 8;
    if (orow < Cout) {
      float v = acc[r] + bias[orow];
      const size_t idx = (size_t)b * Cout * HW + (size_t)orow * HW + col;
      if (res) v += res[idx];
      Y[idx] = act_apply(v, act);
    }
  }
}

// ---------------------------------------------------------------------------
// Fused attention branch.  Layouts straight from the conv tensors:
//   Q: [B][2C][HW]  (branch br, head h, dim d at channel br*C + h*hd + d)
//   K,V: [B][C][NK] (NK = 256 always: KV reduced to 16x16 at every stage)
//   O: [B][2C][HW]  (written at branch offset -> concat for `proj` is free)
// grid = (HW/64, heads, B), block = 128 (4 waves, 16 query rows per wave).
// S = Q*K^T via fp32 WMMA -> LDS; row softmax; O = P*V via fp32 WMMA with
// A-fragments re-read from LDS (C-layout -> A-layout transpose via LDS).
// ---------------------------------------------------------------------------
__global__ __launch_bounds__(128) void attn_kernel(
    const float* __restrict__ Q, const float* __restrict__ K,
    const float* __restrict__ V, float* __restrict__ O, int heads, int hd,
    int HW, int C, int br, float scale) {
  const int NK = 256;
  __shared__ float S[4][16][256];
  const int b = blockIdx.z;
  const int h = blockIdx.y;
  const int wave = threadIdx.x >> 5;
  const int lane = threadIdx.x & 31;
  const int half = lane >> 4, l16 = lane & 15;
  const int q0 = blockIdx.x * 64 + wave * 16;
  const float* qb = Q + (((size_t)b * 2 + br) * heads + h) * (size_t)hd * HW;
  const float* kb = K + ((size_t)b * heads + h) * (size_t)hd * NK;
  const float* vb = V + ((size_t)b * heads + h) * (size_t)hd * NK;
  float* ob = O + (((size_t)b * 2 + br) * heads + h) * (size_t)hd * HW;

  // S = scale * Q * K^T   (hd is a multiple of 4: 16/24/32)
  for (int nt = 0; nt < 16; ++nt) {
    v8f c = {0.f, 0.f, 0.f, 0.f, 0.f, 0.f, 0.f, 0.f};
    for (int kk = 0; kk < hd; kk += 4) {
      const int kq = kk + half * 2;
      v2f a, bf;
      a.x = qb[(size_t)kq * HW + q0 + l16];
      a.y = qb[(size_t)(kq + 1) * HW + q0 + l16];
      bf.x = kb[kq * NK + nt * 16 + l16];
      bf.y = kb[(kq + 1) * NK + nt * 16 + l16];
      c = wmma_f32_16x16x4(a, bf, c);
    }
#pragma unroll
    for (int r = 0; r < 8; ++r)
      S[wave][r + half * 8][nt * 16 + l16] = c[r] * scale;
  }
  __syncthreads();

  if (half == 0) {  // lanes 0..15: one softmax row each
    float* Sr = &S[wave][l16][0];
    float m = -1e30f;
    for (int j = 0; j < NK; ++j) m = fmaxf(m, Sr[j]);
    float s = 0.f;
    for (int j = 0; j < NK; ++j) {
      const float e = expf(Sr[j] - m);
      Sr[j] = e;
      s += e;
    }
    const float inv = 1.f / s;
    for (int j = 0; j < NK; ++j) Sr[j] *= inv;
  }
  __syncthreads();

  // O = P * V   (branchless V loads: clamp column + 0/1 mask, only hd=24
  // actually masks)
  const int dtiles = (hd + 15) >> 4;
  for (int dt = 0; dt < dtiles; ++dt) {
    v8f c = {0.f, 0.f, 0.f, 0.f, 0.f, 0.f, 0.f, 0.f};
    const int d = dt * 16 + l16;
    const int dc = (d < hd) ? d : (hd - 1);
    const float vmask = (d < hd) ? 1.f : 0.f;
    const float* vrow = vb + (size_t)dc * NK + half * 2;
    for (int kk = 0; kk < NK; kk += 4) {
      const int kq = kk + half * 2;
      v2f a, bf;
      a.x = S[wave][l16][kq];
      a.y = S[wave][l16][kq + 1];
      const v2f v2 = *(const v2f*)(vrow + kk);
      bf.x = v2.x * vmask;
      bf.y = v2.y * vmask;
      c = wmma_f32_16x16x4(a, bf, c);
    }
#pragma unroll
    for (int r = 0; r < 8; ++r) {
      const int qrow = q0 + r + half * 8;
      if (d < hd) ob[(size_t)d * HW + qrow] = c[r];
    }
  }
}

// ---------------------------------------------------------------------------
// Generic direct conv (stems, k=s KV-reduction convs, depthwise 3x3).
// ---------------------------------------------------------------------------
__global__ void dirconv_kernel(const float* __restrict__ X,
                               const float* __restrict__ Wt,
                               const float* __restrict__ bias,
                               float* __restrict__ Y, int B, int Cin, int H,
                               int W, int Cout, int OH, int OW, int k,
                               int stride, int pad, int groups, int act) {
  const long long idx = (long long)blockIdx.x * blockDim.x + threadIdx.x;
  const long long total = (long long)B * Cout * OH * OW;
  if (idx >= total) return;
  const int ox = (int)(idx % OW);
  long long t = idx / OW;
  const int oy = (int)(t % OH);
  t /= OH;
  const int co = (int)(t % Cout);
  const int b = (int)(t / Cout);
  const int cpg = Cin / groups;
  const int g = co / (Cout / groups);
  const float* xb = X + ((size_t)b * Cin + (size_t)g * cpg) * H * W;
  const float* wb = Wt + (size_t)co * cpg * k * k;
  float acc = bias[co];
  const int iy0 = oy * stride - pad, ix0 = ox * stride - pad;
  for (int ci = 0; ci < cpg; ++ci)
    for (int ky = 0; ky < k; ++ky) {
      const int iy = iy0 + ky;
      if (iy < 0 || iy >= H) continue;
      for (int kx = 0; kx < k; ++kx) {
        const int ix = ix0 + kx;
        if (ix < 0 || ix >= W) continue;
        acc += xb[((size_t)ci * H + iy) * W + ix] * wb[(ci * k + ky) * k + kx];
      }
    }
  Y[idx] = act_apply(acc, act);
}

// ---------------------------------------------------------------------------
// GroupNorm over (C/groups * HW) per (batch, group), matching jnp reshape.
// grid = (groups, B), block = 256.  Optional fused GELU.
// ---------------------------------------------------------------------------
__global__ __launch_bounds__(256) void gn_kernel(const float* __restrict__ X,
                                                 const float* __restrict__ gamma,
                                                 const float* __restrict__ beta,
                                                 float* __restrict__ Y, int C,
                                                 int HW, int groups, int act) {
  __shared__ float sh[24];
  const int g = blockIdx.x, b = blockIdx.y;
  const int cpg = C / groups;
  const long long n = (long long)cpg * HW;
  const float* xb = X + ((size_t)b * C + (size_t)g * cpg) * HW;
  float* yb = Y + ((size_t)b * C + (size_t)g * cpg) * HW;
  float s = 0.f, s2 = 0.f;
  for (long long i = threadIdx.x; i < n; i += 256) {
    const float v = xb[i];
    s += v;
    s2 += v * v;
  }
  for (int off = 16; off; off >>= 1) {
    s += __shfl_down(s, off);
    s2 += __shfl_down(s2, off);
  }
  const int wave = threadIdx.x >> 5, lane = threadIdx.x & 31;
  if (lane == 0) {
    sh[wave] = s;
    sh[8 + wave] = s2;
  }
  __syncthreads();
  if (threadIdx.x == 0) {
    float ts = 0.f, ts2 = 0.f;
    for (int i = 0; i < 8; ++i) {
      ts += sh[i];
      ts2 += sh[8 + i];
    }
    const float m = ts / (float)n;
    const float var = ts2 / (float)n - m * m;
    sh[16] = m;
    sh[17] = rsqrtf(var + 1e-5f);
  }
  __syncthreads();
  const float m = sh[16], inv = sh[17];
  for (long long i = threadIdx.x; i < n; i += 256) {
    const int c = g * cpg + (int)(i / HW);
    float v = (xb[i] - m) * inv * gamma[c] + beta[c];
    yb[i] = act_apply(v, act);
  }
}

// ---------------------------------------------------------------------------
// Half-pixel bilinear upsample + accumulate (jax.image.resize 'bilinear').
// ---------------------------------------------------------------------------
__global__ void resize_add_kernel(const float* __restrict__ src,
                                  float* __restrict__ dst, int B, int C, int h,
                                  int w, int OH, int OW, int first) {
  const long long idx = (long long)blockIdx.x * blockDim.x + threadIdx.x;
  const long long total = (long long)B * C * OH * OW;
  if (idx >= total) return;
  const int ox = (int)(idx % OW);
  long long t = idx / OW;
  const int oy = (int)(t % OH);
  t /= OH;
  const int c = (int)(t % C);
  const int b = (int)(t / C);
  const float fy = (oy + 0.5f) * ((float)h / OH) - 0.5f;
  const float fx = (ox + 0.5f) * ((float)w / OW) - 0.5f;
  const int y0 = (int)floorf(fy), x0 = (int)floorf(fx);
  const float ay = fy - (float)y0, ax = fx - (float)x0;
  const int y0c = min(max(y0, 0), h - 1), y1c = min(max(y0 + 1, 0), h - 1);
  const int x0c = min(max(x0, 0), w - 1), x1c = min(max(x0 + 1, 0), w - 1);
  const float* sb = src + ((size_t)b * C + c) * h * w;
  const float v =
      (1.f - ay) * ((1.f - ax) * sb[(size_t)y0c * w + x0c] +
                    ax * sb[(size_t)y0c * w + x1c]) +
      ay * ((1.f - ax) * sb[(size_t)y1c * w + x0c] +
            ax * sb[(size_t)y1c * w + x1c]);
  if (first)
    dst[idx] = v;
  else
    dst[idx] += v;
}

// ---------------------------------------------------------------------------
// Host-side launch helpers
// ---------------------------------------------------------------------------
static void gemm1x1(hipStream_t s, const float* X, const float* Wt,
                    const float* bias, const float* res, float* Y, int B,
                    int Cin, int Cout, int HW, int act) {
  dim3 grid((unsigned)(HW / 128), (unsigned)((Cout + 15) / 16), (unsigned)B);
  gemm1x1_kernel<<<grid, 256, 0, s>>>(X, Wt, bias, res, Y, Cin, Cout, HW, act);
}
static void dirconv(hipStream_t s, const float* X, const float* Wt,
                    const float* bias, float* Y, int B, int Cin, int H, int W,
                    int Cout, int OH, int OW, int k, int stride, int pad,
                    int groups, int act) {
  const long long tot = (long long)B * Cout * OH * OW;
  dirconv_kernel<<<(unsigned)((tot + 255) / 256), 256, 0, s>>>(
      X, Wt, bias, Y, B, Cin, H, W, Cout, OH, OW, k, stride, pad, groups, act);
}
static void gnorm(hipStream_t s, const float* X, const float* gamma,
                  const float* beta, float* Y, int B, int C, int HW, int groups,
                  int act) {
  gn_kernel<<<dim3((unsigned)groups, (unsigned)B), 256, 0, s>>>(
      X, gamma, beta, Y, C, HW, groups, act);
}
static void attn(hipStream_t s, const float* Q, const float* K, const float* V,
                 float* O, int B, int heads, int hd, int HW, int C, int br) {
  const float scale = 1.0f / sqrtf((float)hd);
  attn_kernel<<<dim3((unsigned)(HW / 64), (unsigned)heads, (unsigned)B), 128, 0,
                s>>>(Q, K, V, O, heads, hd, HW, C, br, scale);
}
static void resize_add(hipStream_t s, const float* src, float* dst, int B,
                       int C, int h, int w, int first) {
  const long long tot = (long long)B * C * 512 * 512;
  resize_add_kernel<<<(unsigned)((tot + 255) / 256), 256, 0, s>>>(
      src, dst, B, C, h, w, 512, 512, first);
}

// ---------------------------------------------------------------------------
// Full forward pass.  Param indices follow jax tree_flatten (sorted dict
// keys): [0]=x, then dec.seg{b,w}, dec.up[i]{conv{b,w},gn{b,g}}, then
// stages[i]{ blocks[j]{dw,fc1,fc2,n1,n2,proj,q,spak_dw,spak_pw,spav_dw,
// spav_pw,spek,spev,stride (each {b,w}/{b,g})}, norm{b,g}, pe{conv{b,w},
// gn{b,g}} }.
// ---------------------------------------------------------------------------
extern "C" void kernel_launch(void* const* d_in, const int* in_sizes, int n_in,
                              void* d_out, int out_size, void* d_ws,
                              size_t ws_size, hipStream_t stream) {
  (void)in_sizes; (void)n_in; (void)out_size; (void)ws_size;
  const float* x_in = (const float*)d_in[0];
  float* ws = (float*)d_ws;
  float* out = (float*)d_out;
  auto P = [&](int i) { return (const float*)d_in[i]; };

  const int B = 4;
  const int CH[4] = {32, 64, 96, 128};
  const int HEADS_[4] = {1, 2, 4, 8};
  const int STR_[4] = {8, 4, 2, 1};
  const int MLPR[4] = {8, 8, 4, 4};
  const int NG[4] = {1, 2, 4, 8};
  const int HS[4] = {128, 64, 32, 16};

  // scratch arena (float offsets)
  size_t off = 0;
  auto alloc = [&](size_t n) {
    const size_t o = off;
    off += (n + 63) & ~(size_t)63;
    return o;
  };
  const size_t ACC = alloc((size_t)4 * 32 * 512 * 512);
  const size_t HID1 = alloc(16777216);
  const size_t HID2 = alloc(16777216);
  const size_t XBUF = alloc(2097152);
  const size_t TBUF = alloc(2097152);
  const size_t QBUF = alloc(4194304);
  const size_t ABUF = alloc(4194304);
  const size_t XS = alloc(131072);
  const size_t KV0 = alloc(131072);
  const size_t KA = alloc(131072);
  const size_t VA = alloc(131072);
  const size_t KE = alloc(131072);
  const size_t VE = alloc(131072);
  size_t OUTS[4];
  for (int i = 0; i < 4; ++i)
    OUTS[i] = alloc((size_t)B * CH[i] * HS[i] * HS[i]);
  const size_t DECT = alloc(2097152);

  // ---- backbone ----
  const float* cur = x_in;
  int inC = 32, inH = 512, inW = 512;
  for (int i = 0; i < 4; ++i) {
    const int C = CH[i];
    const int k = (i == 0) ? 7 : 3, s = (i == 0) ? 4 : 2;
    const int H = inH / s, W = inW / s, HW = H * W;
    const int pe = 19 + 62 * i + 58;  // conv.b, conv.w, gn.b, gn.g
    dirconv(stream, cur, P(pe + 1), P(pe + 0), ws + TBUF, B, inC, inH, inW, C,
            H, W, k, s, k / 2, 1, 0);
    gnorm(stream, ws + TBUF, P(pe + 3), P(pe + 2), ws + XBUF, B, C, HW, C / 32,
          0);

    const int heads = HEADS_[i], hd = C / heads, ss = STR_[i];
    const int hid = C * MLPR[i];
    for (int j = 0; j < 2; ++j) {
      const int bb = 19 + 62 * i + 28 * j;
      // ---- SSAA attention ----
      gnorm(stream, ws + XBUF, P(bb + 7), P(bb + 6), ws + TBUF, B, C, HW, 4, 0);
      gemm1x1(stream, ws + TBUF, P(bb + 13), P(bb + 12), nullptr, ws + QBUF, B,
              C, 2 * C, HW, 0);
      dirconv(stream, ws + TBUF, P(bb + 27), P(bb + 26), ws + XS, B, C, H, W, C,
              16, 16, ss, ss, 0, 1, 0);  // strided KV reduction -> 16x16
      dirconv(stream, ws + XS, P(bb + 15), P(bb + 14), ws + KV0, B, C, 16, 16,
              C, 16, 16, 3, 1, 1, C, 0);  // spak_dw
      gemm1x1(stream, ws + KV0, P(bb + 17), P(bb + 16), nullptr, ws + KA, B, C,
              C, 256, 0);  // spak_pw
      dirconv(stream, ws + XS, P(bb + 19), P(bb + 18), ws + KV0, B, C, 16, 16,
              C, 16, 16, 3, 1, 1, C, 0);  // spav_dw
      gemm1x1(stream, ws + KV0, P(bb + 21), P(bb + 20), nullptr, ws + VA, B, C,
              C, 256, 0);  // spav_pw
      gemm1x1(stream, ws + XS, P(bb + 23), P(bb + 22), nullptr, ws + KE, B, C,
              C, 256, 0);  // spek
      gemm1x1(stream, ws + XS, P(bb + 25), P(bb + 24), nullptr, ws + VE, B, C,
              C, 256, 0);  // spev
      attn(stream, ws + QBUF, ws + KA, ws + VA, ws + ABUF, B, heads, hd, HW, C,
           0);
      attn(stream, ws + QBUF, ws + KE, ws + VE, ws + ABUF, B, heads, hd, HW, C,
           1);
      gemm1x1(stream, ws + ABUF, P(bb + 11), P(bb + 10), ws + XBUF, ws + XBUF,
              B, 2 * C, C, HW, 0);  // proj + residual
      // ---- MLP ----
      gnorm(stream, ws + XBUF, P(bb + 9), P(bb + 8), ws + TBUF, B, C, HW, 4, 0);
      gemm1x1(stream, ws + TBUF, P(bb + 3), P(bb + 2), nullptr, ws + HID1, B, C,
              hid, HW, 0);  // fc1
      dirconv(stream, ws + HID1, P(bb + 1), P(bb + 0), ws + HID2, B, hid, H, W,
              hid, H, W, 3, 1, 1, hid, 1);  // dw + gelu
      gemm1x1(stream, ws + HID2, P(bb + 5), P(bb + 4), ws + XBUF, ws + XBUF, B,
              hid, C, HW, 0);  // fc2 + residual
    }
    const int nb = 19 + 62 * i + 56;
    gnorm(stream, ws + XBUF, P(nb + 1), P(nb + 0), ws + OUTS[i], B, C, HW,
          NG[i], 0);
    cur = ws + OUTS[i];
    inC = C;
    inH = H;
    inW = W;
  }

  // ---- decoder ----
  for (int i = 0; i < 4; ++i) {
    const int ub = 3 + 4 * i;  // conv.b, conv.w, gn.b, gn.g
    const int hw = HS[i] * HS[i];
    gemm1x1(stream, ws + OUTS[i], P(ub + 1), P(ub + 0), nullptr, ws + DECT, B,
            CH[i], 32, hw, 0);
    gnorm(stream, ws + DECT, P(ub + 3), P(ub + 2), ws + DECT, B, 32, hw, 1, 1);
    resize_add(stream, ws + DECT, ws + ACC, B, 32, HS[i], HS[i], i == 0);
  }
  // seg 1x1 conv (Cout=1, masked M tile) + fused sigmoid -> d_out
  gemm1x1(stream, ws + ACC, P(2), P(1), nullptr, out, B, 32, 1, 512 * 512, 2);
}